// Mean_module_28595892257584
// MI455X (gfx1250) — compile-verified
//
#include <hip/hip_runtime.h>
#include <hip/hip_bf16.h>
#include <stdint.h>

// Problem constants (from the reference)
#define N_AGENTS 32768
#define FEAT     64     // D: reduced dimension
#define FRAMES   64     // T: kept dimension
#define WAVES_PER_BLOCK 2
#define ROW_DW   72     // padded LDS row stride (dwords): 2*72 % 64 == 16 -> halves hit disjoint banks

typedef __attribute__((ext_vector_type(2))) float v2f;
typedef __attribute__((ext_vector_type(8))) float v8f;

// out[n,d,t] = (1/64) * sum_{d'} in[n,d',t], broadcast over d.
// One wave handles one agent (a 64x64 f32 tile).
__global__ __launch_bounds__(WAVES_PER_BLOCK * 32)
void mean_bcast_wmma(const float* __restrict__ in, float* __restrict__ out) {
    __shared__ __align__(16) float tile[WAVES_PER_BLOCK][FEAT * ROW_DW]; // 2 * 18 KB
    __shared__ __align__(16) float mrow[WAVES_PER_BLOCK][FRAMES];        // 2 * 256 B

    const int lane  = threadIdx.x & 31;
    const int wave  = threadIdx.x >> 5;
    const int n     = blockIdx.x * WAVES_PER_BLOCK + wave;   // grid exactly covers 32768 agents

    const float* __restrict__ gsrc = in  + (size_t)n * (FEAT * FRAMES);
    float*       __restrict__ gdst = out + (size_t)n * (FEAT * FRAMES);

    const int c4    = lane & 15;   // 16-byte column segment within a 256 B row
    const int rhalf = lane >> 4;   // 0 or 1: which of the 2 rows this half-wave covers

    // ---- Stage 1: async-copy the 16 KB agent tile global -> LDS, b128 per lane
    // (512 B contiguous per instruction = 2 full input rows), padded LDS rows.
    #pragma unroll
    for (int i = 0; i < 32; ++i) {
        const int row = 2 * i + rhalf;
        uint64_t gaddr = (uint64_t)(uintptr_t)(gsrc + row * FRAMES + c4 * 4);
        uint32_t laddr = (uint32_t)(uintptr_t)(&tile[wave][row * ROW_DW + c4 * 4]);
        asm volatile("global_load_async_to_lds_b128 %0, %1, off"
                     :: "v"(laddr), "v"(gaddr)
                     : "memory");
    }
    asm volatile("s_wait_asynccnt 0x0" ::: "memory");

    // ---- Stage 2: WMMA reduction over D with all-ones A.
    // C[m,t] = sum_k B[k,t]; every row of C is the column sum (broadcast for free).
    v2f ones; ones.x = 1.0f; ones.y = 1.0f;
    const int tcol = lane & 15;
    const int rofs = (lane >> 4) << 1;   // half-wave 0 supplies rows {4kc,4kc+1}, half 1 {4kc+2,4kc+3}

    #pragma unroll
    for (int tt = 0; tt < 4; ++tt) {
        const int t0 = tt * 16;
        v8f acc = {};
        #pragma unroll
        for (int kc = 0; kc < 16; ++kc) {
            const int r0 = 4 * kc + rofs;
            v2f b;
            b.x = tile[wave][(r0    ) * ROW_DW + t0 + tcol];
            b.y = tile[wave][(r0 + 1) * ROW_DW + t0 + tcol];
            // D = A(ones,16x4) * B(4x16 slab of D x T) + C   (f32-exact accumulate)
            acc = __builtin_amdgcn_wmma_f32_16x16x4_f32(
                      false, ones, false, b, (short)0, acc, false, false);
        }
        const float mean = acc[0] * (1.0f / FEAT);  // lane L holds t = t0 + (L & 15)
        if (lane < 16) mrow[wave][t0 + lane] = mean;
    }
    __syncthreads();

    // ---- Stage 3: broadcast-store. Each lane re-reads its float4 of means from LDS,
    // then 32 b128 stores write 2 identical full output rows per instruction.
    const float4 v = *(const float4*)&mrow[wave][4 * (lane & 15)];
    #pragma unroll
    for (int i = 0; i < 32; ++i) {
        const int row = 2 * i + rhalf;
        *(float4*)(gdst + row * FRAMES + 4 * (lane & 15)) = v;
    }
}

extern "C" void kernel_launch(void* const* d_in, const int* in_sizes, int n_in,
                              void* d_out, int out_size, void* d_ws, size_t ws_size,
                              hipStream_t stream) {
    (void)in_sizes; (void)n_in; (void)out_size; (void)d_ws; (void)ws_size;
    const float* in  = (const float*)d_in[0];   // in_features [32768, 64, 64] f32
    float*       out = (float*)d_out;           // [32768, 64, 64] f32
    dim3 grid(N_AGENTS / WAVES_PER_BLOCK);
    dim3 block(WAVES_PER_BLOCK * 32);
    hipLaunchKernelGGL(mean_bcast_wmma, grid, block, 0, stream, in, out);
}